// PeriodicSetTransformer_45337674776590
// MI455X (gfx1250) — compile-verified
//
#include <hip/hip_runtime.h>
#include <math.h>
#include <stdint.h>

typedef float v2f __attribute__((ext_vector_type(2)));
typedef float v8f __attribute__((ext_vector_type(8)));

#define DEVINL static __device__ __forceinline__

constexpr int BB   = 64;    // batch
constexpr int NN   = 512;   // set size
constexpr int FD   = 101;   // str features
constexpr int EE   = 128;   // embed dim
constexpr int DD2  = 64;    // q/k dim
constexpr int AFD  = 200;   // atom feature dim
constexpr float LNEPS = 1e-5f;

// ---------------- WMMA helper: D = A(16x4,f32) * B(4x16,f32) + C ------------
DEVINL v8f wmma4(v2f a, v2f b, v8f c) {
  return __builtin_amdgcn_wmma_f32_16x16x4_f32(false, a, false, b, (short)0, c,
                                               false, false);
}

// ---------------- CDNA5 async global->LDS staging (ASYNCcnt) ----------------
DEVINL void async_ld32(uint32_t lds_off, const void* gaddr) {
  asm volatile("global_load_async_to_lds_b32 %0, %1, off"
               :: "v"(lds_off), "v"((uint64_t)(uintptr_t)gaddr) : "memory");
}
DEVINL void async_wait0() {
  asm volatile("s_wait_asynccnt 0" ::: "memory");
}

// ---------------- embedding: x = gather@comp_W + polyexpand@pdd_W -----------
__global__ void embed_kernel(const float* __restrict__ str_fea,
                             const int* __restrict__ comp_fea,
                             const float* __restrict__ atom_table,
                             const float* __restrict__ comp_W,
                             const float* __restrict__ comp_b,
                             const float* __restrict__ pdd_W,
                             const float* __restrict__ pdd_b,
                             float* __restrict__ x, float* __restrict__ x_init) {
  __shared__ float sAtom[AFD];
  __shared__ float sD[FD - 1];
  int row = blockIdx.x;            // b*N + n
  int e   = threadIdx.x;           // 0..127
  int idx = comp_fea[row];
  const float* arow = atom_table + (size_t)idx * AFD;
  const float* srow = str_fea + (size_t)row * FD;
  // async DMA-style fill of LDS (GLOBAL_LOAD_ASYNC_TO_LDS_B32, ASYNCcnt)
  async_ld32((uint32_t)(uintptr_t)&sAtom[e], arow + e);
  if (e < AFD - EE) async_ld32((uint32_t)(uintptr_t)&sAtom[EE + e], arow + EE + e);
  if (e < FD - 1)   async_ld32((uint32_t)(uintptr_t)&sD[e], srow + 1 + e);
  async_wait0();
  __syncthreads();
  float acc = comp_b[e] + pdd_b[e];
  for (int k = 0; k < AFD; ++k) acc += sAtom[k] * comp_W[k * EE + e];
  for (int f = 0; f < FD - 1; ++f) {
    float dv = sD[f];
#pragma unroll
    for (int j = 0; j < 10; ++j) {
      float t = 1.0f - (dv - (float)j * 0.1f);
      acc += t * t * pdd_W[(f * 10 + j) * EE + e];
    }
  }
  size_t o = (size_t)row * EE + e;
  x[o] = acc;
  x_init[o] = acc;
}

// ---------------- LayerNorm over E, optional transpose to (B,E,N) -----------
template <bool TRANSPOSE>
__global__ void ln_kernel(const float* __restrict__ in, const float* __restrict__ g,
                          const float* __restrict__ bt, float* __restrict__ out) {
  __shared__ float sh[EE];
  int row = blockIdx.x;            // b*N + n
  int e   = threadIdx.x;
  float v = in[(size_t)row * EE + e];
  sh[e] = v; __syncthreads();
  for (int s = EE / 2; s > 0; s >>= 1) { if (e < s) sh[e] += sh[e + s]; __syncthreads(); }
  float mean = sh[0] * (1.0f / EE); __syncthreads();
  float d = v - mean;
  sh[e] = d * d; __syncthreads();
  for (int s = EE / 2; s > 0; s >>= 1) { if (e < s) sh[e] += sh[e + s]; __syncthreads(); }
  float var = sh[0] * (1.0f / EE); __syncthreads();
  float r = (v - mean) * rsqrtf(var + LNEPS) * g[e] + bt[e];
  if (TRANSPOSE) {
    int b = row >> 9, n = row & (NN - 1);
    out[((size_t)(b * EE + e)) * NN + n] = r;
  } else {
    out[(size_t)row * EE + e] = r;
  }
}

// ---------------- q = h^T(NxC) x qW^T(Cx64), 32x32 wave tiles ---------------
__global__ void gemm_q(const float* __restrict__ h, const float* __restrict__ qW,
                       float* __restrict__ q) {
  int lane = threadIdx.x & 31;
  int gt = blockIdx.x * 4 + (threadIdx.x >> 5);
  int b = gt >> 5, t = gt & 31;            // (512/32)*(64/32)=32 tiles per batch
  int n0 = (t >> 1) * 32, d0 = (t & 1) * 32;
  const float* hb = h + (size_t)b * EE * NN;
  int ml = lane & 15, kh = (lane >> 4) * 2;
  v8f c00 = {}, c01 = {}, c10 = {}, c11 = {};
  for (int k0 = 0; k0 < EE; k0 += 4) {
    int kk = k0 + kh;
    const float* h0 = hb + (size_t)kk * NN;
    v2f a0, a1, b0, b1;
    a0.x = h0[n0 + ml];      a0.y = h0[NN + n0 + ml];
    a1.x = h0[n0 + 16 + ml]; a1.y = h0[NN + n0 + 16 + ml];
    b0.x = qW[(d0 + ml) * EE + kk];      b0.y = qW[(d0 + ml) * EE + kk + 1];
    b1.x = qW[(d0 + 16 + ml) * EE + kk]; b1.y = qW[(d0 + 16 + ml) * EE + kk + 1];
    c00 = wmma4(a0, b0, c00);
    c01 = wmma4(a0, b1, c01);
    c10 = wmma4(a1, b0, c10);
    c11 = wmma4(a1, b1, c11);
  }
#pragma unroll
  for (int i = 0; i < 8; ++i) {
    int rit = i + (lane >> 4) * 8;
    size_t r0 = ((size_t)b * NN + n0 + rit) * DD2;
    size_t r1 = ((size_t)b * NN + n0 + 16 + rit) * DD2;
    q[r0 + d0 + ml]      = c00[i];
    q[r0 + d0 + 16 + ml] = c01[i];
    q[r1 + d0 + ml]      = c10[i];
    q[r1 + d0 + 16 + ml] = c11[i];
  }
}

// ---------------- k/v = W(DRxC) x h(CxN) + bias -> (B,DR,N) -----------------
template <int DR>
__global__ void gemm_proj(const float* __restrict__ h, const float* __restrict__ W,
                          const float* __restrict__ bias, float* __restrict__ out) {
  int lane = threadIdx.x & 31;
  int gt = blockIdx.x * 4 + (threadIdx.x >> 5);
  constexpr int TPB = (DR / 32) * (NN / 32);
  int b = gt / TPB, t = gt % TPB;
  int d0 = (t / (NN / 32)) * 32, n0 = (t % (NN / 32)) * 32;
  const float* hb = h + (size_t)b * EE * NN;
  int ml = lane & 15, kh = (lane >> 4) * 2;
  v8f c00 = {}, c01 = {}, c10 = {}, c11 = {};
  for (int k0 = 0; k0 < EE; k0 += 4) {
    int kk = k0 + kh;
    const float* h0 = hb + (size_t)kk * NN;
    v2f a0, a1, b0, b1;
    a0.x = W[(d0 + ml) * EE + kk];      a0.y = W[(d0 + ml) * EE + kk + 1];
    a1.x = W[(d0 + 16 + ml) * EE + kk]; a1.y = W[(d0 + 16 + ml) * EE + kk + 1];
    b0.x = h0[n0 + ml];      b0.y = h0[NN + n0 + ml];
    b1.x = h0[n0 + 16 + ml]; b1.y = h0[NN + n0 + 16 + ml];
    c00 = wmma4(a0, b0, c00);
    c01 = wmma4(a0, b1, c01);
    c10 = wmma4(a1, b0, c10);
    c11 = wmma4(a1, b1, c11);
  }
#pragma unroll
  for (int i = 0; i < 8; ++i) {
    int rit = i + (lane >> 4) * 8;
    int r0 = d0 + rit, r1 = d0 + 16 + rit;
    float bz0 = bias ? bias[r0] : 0.0f;
    float bz1 = bias ? bias[r1] : 0.0f;
    out[((size_t)b * DR + r0) * NN + n0 + ml]      = c00[i] + bz0;
    out[((size_t)b * DR + r0) * NN + n0 + 16 + ml] = c01[i] + bz0;
    out[((size_t)b * DR + r1) * NN + n0 + ml]      = c10[i] + bz1;
    out[((size_t)b * DR + r1) * NN + n0 + 16 + ml] = c11[i] + bz1;
  }
}

// ---------------- energy = q(Nx64) x k(64xN) -> attn(B,N,N) -----------------
__global__ void gemm_energy(const float* __restrict__ q, const float* __restrict__ kb,
                            float* __restrict__ attn) {
  int lane = threadIdx.x & 31;
  int gt = blockIdx.x * 4 + (threadIdx.x >> 5);
  int b = gt >> 8, t = gt & 255;           // 16*16 tiles per batch
  int n0 = (t >> 4) * 32, m0 = (t & 15) * 32;
  const float* qb  = q  + (size_t)b * NN * DD2;
  const float* kbb = kb + (size_t)b * DD2 * NN;
  int ml = lane & 15, kh = (lane >> 4) * 2;
  v8f c00 = {}, c01 = {}, c10 = {}, c11 = {};
  for (int k0 = 0; k0 < DD2; k0 += 4) {
    int kk = k0 + kh;
    const float* kp = kbb + (size_t)kk * NN;
    v2f a0, a1, b0, b1;
    a0.x = qb[(size_t)(n0 + ml) * DD2 + kk];      a0.y = qb[(size_t)(n0 + ml) * DD2 + kk + 1];
    a1.x = qb[(size_t)(n0 + 16 + ml) * DD2 + kk]; a1.y = qb[(size_t)(n0 + 16 + ml) * DD2 + kk + 1];
    b0.x = kp[m0 + ml];      b0.y = kp[NN + m0 + ml];
    b1.x = kp[m0 + 16 + ml]; b1.y = kp[NN + m0 + 16 + ml];
    c00 = wmma4(a0, b0, c00);
    c01 = wmma4(a0, b1, c01);
    c10 = wmma4(a1, b0, c10);
    c11 = wmma4(a1, b1, c11);
  }
#pragma unroll
  for (int i = 0; i < 8; ++i) {
    int rit = i + (lane >> 4) * 8;
    float* p0 = attn + ((size_t)b * NN + n0 + rit) * NN;
    float* p1 = attn + ((size_t)b * NN + n0 + 16 + rit) * NN;
    p0[m0 + ml]      = c00[i];
    p0[m0 + 16 + ml] = c01[i];
    p1[m0 + ml]      = c10[i];
    p1[m0 + 16 + ml] = c11[i];
  }
}

// ---------------- row softmax over last axis (length 512) -------------------
__global__ void softmax_rows(float* __restrict__ attn) {
  __shared__ float sh[256];
  int row = blockIdx.x, tid = threadIdx.x;
  float* p = attn + (size_t)row * NN;
  float v0 = p[tid], v1 = p[tid + 256];
  sh[tid] = fmaxf(v0, v1); __syncthreads();
  for (int s = 128; s > 0; s >>= 1) { if (tid < s) sh[tid] = fmaxf(sh[tid], sh[tid + s]); __syncthreads(); }
  float mx = sh[0]; __syncthreads();
  v0 = expf(v0 - mx); v1 = expf(v1 - mx);
  sh[tid] = v0 + v1; __syncthreads();
  for (int s = 128; s > 0; s >>= 1) { if (tid < s) sh[tid] += sh[tid + s]; __syncthreads(); }
  float inv = 1.0f / sh[0];
  p[tid] = v0 * inv; p[tid + 256] = v1 * inv;
}

// ---------------- column sums over axis n (for the second normalization) ----
__global__ void colsum_kernel(const float* __restrict__ attn, float* __restrict__ cs) {
  int b = blockIdx.x >> 1;
  int m = ((blockIdx.x & 1) << 8) + threadIdx.x;
  const float* p = attn + (size_t)b * NN * NN + m;
  float s = 0.0f;
  for (int n = 0; n < NN; ++n) s += p[(size_t)n * NN];
  cs[b * NN + m] = s;
}

// ---------------- x_r = v(ExN) x attn(NxN), scaled by 1/colsum --------------
__global__ void gemm_xr(const float* __restrict__ v, const float* __restrict__ attn,
                        const float* __restrict__ cs, float* __restrict__ xr) {
  int lane = threadIdx.x & 31;
  int gt = blockIdx.x * 4 + (threadIdx.x >> 5);
  int b = gt >> 6, t = gt & 63;            // 4*16 tiles per batch
  int c0 = (t >> 4) * 32, m0 = (t & 15) * 32;
  const float* vb = v + (size_t)b * EE * NN;
  const float* ab = attn + (size_t)b * NN * NN;
  int ml = lane & 15, kh = (lane >> 4) * 2;
  v8f c00 = {}, c01 = {}, c10 = {}, c11 = {};
  for (int k0 = 0; k0 < NN; k0 += 4) {
    int kk = k0 + kh;
    const float* ap = ab + (size_t)kk * NN;
    v2f a0, a1, b0, b1;
    a0.x = vb[(size_t)(c0 + ml) * NN + kk];      a0.y = vb[(size_t)(c0 + ml) * NN + kk + 1];
    a1.x = vb[(size_t)(c0 + 16 + ml) * NN + kk]; a1.y = vb[(size_t)(c0 + 16 + ml) * NN + kk + 1];
    b0.x = ap[m0 + ml];      b0.y = ap[NN + m0 + ml];
    b1.x = ap[m0 + 16 + ml]; b1.y = ap[NN + m0 + 16 + ml];
    c00 = wmma4(a0, b0, c00);
    c01 = wmma4(a0, b1, c01);
    c10 = wmma4(a1, b0, c10);
    c11 = wmma4(a1, b1, c11);
  }
  float s0 = 1.0f / (1e-9f + cs[b * NN + m0 + ml]);
  float s1 = 1.0f / (1e-9f + cs[b * NN + m0 + 16 + ml]);
#pragma unroll
  for (int i = 0; i < 8; ++i) {
    int rit = i + (lane >> 4) * 8;
    float* p0 = xr + ((size_t)b * EE + c0 + rit) * NN;
    float* p1 = xr + ((size_t)b * EE + c0 + 16 + rit) * NN;
    p0[m0 + ml]      = c00[i] * s0;
    p0[m0 + 16 + ml] = c01[i] * s1;
    p1[m0 + ml]      = c10[i] * s0;
    p1[m0 + 16 + ml] = c11[i] * s1;
  }
}

// ---------------- t = tW x (sain - x_r); out = sain + relu(bn(t)) -----------
__global__ void gemm_t(const float* __restrict__ sain, const float* __restrict__ xr,
                       const float* __restrict__ tW, const float* __restrict__ tb,
                       const float* __restrict__ bg, const float* __restrict__ bb,
                       float* __restrict__ out) {
  int lane = threadIdx.x & 31;
  int gt = blockIdx.x * 4 + (threadIdx.x >> 5);
  int b = gt >> 6, t = gt & 63;
  int d0 = (t >> 4) * 32, n0 = (t & 15) * 32;
  const float* sb = sain + (size_t)b * EE * NN;
  const float* xb = xr + (size_t)b * EE * NN;
  int ml = lane & 15, kh = (lane >> 4) * 2;
  v8f c00 = {}, c01 = {}, c10 = {}, c11 = {};
  for (int k0 = 0; k0 < EE; k0 += 4) {
    int kk = k0 + kh;
    size_t r0 = (size_t)kk * NN, r1 = (size_t)(kk + 1) * NN;
    v2f a0, a1, b0, b1;
    a0.x = tW[(d0 + ml) * EE + kk];      a0.y = tW[(d0 + ml) * EE + kk + 1];
    a1.x = tW[(d0 + 16 + ml) * EE + kk]; a1.y = tW[(d0 + 16 + ml) * EE + kk + 1];
    b0.x = sb[r0 + n0 + ml]      - xb[r0 + n0 + ml];
    b0.y = sb[r1 + n0 + ml]      - xb[r1 + n0 + ml];
    b1.x = sb[r0 + n0 + 16 + ml] - xb[r0 + n0 + 16 + ml];
    b1.y = sb[r1 + n0 + 16 + ml] - xb[r1 + n0 + 16 + ml];
    c00 = wmma4(a0, b0, c00);
    c01 = wmma4(a0, b1, c01);
    c10 = wmma4(a1, b0, c10);
    c11 = wmma4(a1, b1, c11);
  }
  const float invs = rsqrtf(1.0f + LNEPS);
#pragma unroll
  for (int i = 0; i < 8; ++i) {
    int rit = i + (lane >> 4) * 8;
    int dA = d0 + rit, dB = d0 + 16 + rit;
    float vA0 = (c00[i] + tb[dA]) * invs * bg[dA] + bb[dA];
    float vA1 = (c01[i] + tb[dA]) * invs * bg[dA] + bb[dA];
    float vB0 = (c10[i] + tb[dB]) * invs * bg[dB] + bb[dB];
    float vB1 = (c11[i] + tb[dB]) * invs * bg[dB] + bb[dB];
    size_t oA = ((size_t)b * EE + dA) * NN, oB = ((size_t)b * EE + dB) * NN;
    out[oA + n0 + ml]      = sb[(size_t)dA * NN + n0 + ml]      + fmaxf(vA0, 0.0f);
    out[oA + n0 + 16 + ml] = sb[(size_t)dA * NN + n0 + 16 + ml] + fmaxf(vA1, 0.0f);
    out[oB + n0 + ml]      = sb[(size_t)dB * NN + n0 + ml]      + fmaxf(vB0, 0.0f);
    out[oB + n0 + 16 + ml] = sb[(size_t)dB * NN + n0 + 16 + ml] + fmaxf(vB1, 0.0f);
  }
}

// ---------------- out1 = x + [x1;x2]^T(Nx2E) x W(2ExE) + bias ---------------
__global__ void gemm_out1(const float* __restrict__ x1, const float* __restrict__ x2,
                          const float* __restrict__ x, const float* __restrict__ W,
                          const float* __restrict__ bias, float* __restrict__ out1) {
  int lane = threadIdx.x & 31;
  int gt = blockIdx.x * 4 + (threadIdx.x >> 5);
  int b = gt >> 6, t = gt & 63;            // 16*4 tiles per batch
  int n0 = (t >> 2) * 32, e0 = (t & 3) * 32;
  const float* x1b = x1 + (size_t)b * EE * NN;
  const float* x2b = x2 + (size_t)b * EE * NN;
  int ml = lane & 15, kh = (lane >> 4) * 2;
  v8f c00 = {}, c01 = {}, c10 = {}, c11 = {};
  for (int k0 = 0; k0 < 2 * EE; k0 += 4) {
    int kk = k0 + kh;
    const float* src = (kk < EE) ? x1b : x2b;
    int kc = kk & (EE - 1);
    const float* s0 = src + (size_t)kc * NN;
    v2f a0, a1, b0, b1;
    a0.x = s0[n0 + ml];      a0.y = s0[NN + n0 + ml];
    a1.x = s0[n0 + 16 + ml]; a1.y = s0[NN + n0 + 16 + ml];
    b0.x = W[kk * EE + e0 + ml];      b0.y = W[(kk + 1) * EE + e0 + ml];
    b1.x = W[kk * EE + e0 + 16 + ml]; b1.y = W[(kk + 1) * EE + e0 + 16 + ml];
    c00 = wmma4(a0, b0, c00);
    c01 = wmma4(a0, b1, c01);
    c10 = wmma4(a1, b0, c10);
    c11 = wmma4(a1, b1, c11);
  }
#pragma unroll
  for (int i = 0; i < 8; ++i) {
    int rit = i + (lane >> 4) * 8;
    size_t oA = ((size_t)b * NN + n0 + rit) * EE;
    size_t oB = ((size_t)b * NN + n0 + 16 + rit) * EE;
    out1[oA + e0 + ml]      = c00[i] + x[oA + e0 + ml]      + bias[e0 + ml];
    out1[oA + e0 + 16 + ml] = c01[i] + x[oA + e0 + 16 + ml] + bias[e0 + 16 + ml];
    out1[oB + e0 + ml]      = c10[i] + x[oB + e0 + ml]      + bias[e0 + ml];
    out1[oB + e0 + 16 + ml] = c11[i] + x[oB + e0 + 16 + ml] + bias[e0 + 16 + ml];
  }
}

// ---------------- s = out1 + mish(xn x W + bias) ----------------------------
DEVINL float mishf(float z) {
  float sp = (z > 20.0f) ? z : log1pf(expf(z));
  return z * tanhf(sp);
}

__global__ void gemm_ffn(const float* __restrict__ xn, const float* __restrict__ out1,
                         const float* __restrict__ W, const float* __restrict__ bias,
                         float* __restrict__ s) {
  int lane = threadIdx.x & 31;
  int gt = blockIdx.x * 4 + (threadIdx.x >> 5);
  int b = gt >> 6, t = gt & 63;
  int n0 = (t >> 2) * 32, e0 = (t & 3) * 32;
  const float* xb = xn + (size_t)b * NN * EE;
  int ml = lane & 15, kh = (lane >> 4) * 2;
  v8f c00 = {}, c01 = {}, c10 = {}, c11 = {};
  for (int k0 = 0; k0 < EE; k0 += 4) {
    int kk = k0 + kh;
    v2f a0, a1, b0, b1;
    a0.x = xb[(size_t)(n0 + ml) * EE + kk];      a0.y = xb[(size_t)(n0 + ml) * EE + kk + 1];
    a1.x = xb[(size_t)(n0 + 16 + ml) * EE + kk]; a1.y = xb[(size_t)(n0 + 16 + ml) * EE + kk + 1];
    b0.x = W[kk * EE + e0 + ml];      b0.y = W[(kk + 1) * EE + e0 + ml];
    b1.x = W[kk * EE + e0 + 16 + ml]; b1.y = W[(kk + 1) * EE + e0 + 16 + ml];
    c00 = wmma4(a0, b0, c00);
    c01 = wmma4(a0, b1, c01);
    c10 = wmma4(a1, b0, c10);
    c11 = wmma4(a1, b1, c11);
  }
#pragma unroll
  for (int i = 0; i < 8; ++i) {
    int rit = i + (lane >> 4) * 8;
    size_t oA = ((size_t)b * NN + n0 + rit) * EE;
    size_t oB = ((size_t)b * NN + n0 + 16 + rit) * EE;
    s[oA + e0 + ml]      = out1[oA + e0 + ml]      + mishf(c00[i] + bias[e0 + ml]);
    s[oA + e0 + 16 + ml] = out1[oA + e0 + 16 + ml] + mishf(c01[i] + bias[e0 + 16 + ml]);
    s[oB + e0 + ml]      = out1[oB + e0 + ml]      + mishf(c10[i] + bias[e0 + ml]);
    s[oB + e0 + 16 + ml] = out1[oB + e0 + 16 + ml] + mishf(c11[i] + bias[e0 + 16 + ml]);
  }
}

// ---------------- weighted pool + LN + final dot ----------------------------
__global__ void pool_kernel(const float* __restrict__ x, const float* __restrict__ x_init,
                            const float* __restrict__ str_fea,
                            const float* __restrict__ g2, const float* __restrict__ b2,
                            const float* __restrict__ fW, const float* __restrict__ fb,
                            float* __restrict__ out) {
  __shared__ float sh[EE];
  int b = blockIdx.x, e = threadIdx.x;
  float acc = 0.0f;
  for (int n = 0; n < NN; ++n) {
    size_t r = (size_t)b * NN + n;
    float w = str_fea[r * FD];
    acc += w * (x[r * EE + e] + x_init[r * EE + e]);
  }
  sh[e] = acc; __syncthreads();
  for (int s = EE / 2; s > 0; s >>= 1) { if (e < s) sh[e] += sh[e + s]; __syncthreads(); }
  float mean = sh[0] * (1.0f / EE); __syncthreads();
  float d = acc - mean;
  sh[e] = d * d; __syncthreads();
  for (int s = EE / 2; s > 0; s >>= 1) { if (e < s) sh[e] += sh[e + s]; __syncthreads(); }
  float var = sh[0] * (1.0f / EE); __syncthreads();
  float xv = (acc - mean) * rsqrtf(var + LNEPS) * g2[e] + b2[e];
  sh[e] = xv * fW[e]; __syncthreads();
  for (int s = EE / 2; s > 0; s >>= 1) { if (e < s) sh[e] += sh[e + s]; __syncthreads(); }
  if (e == 0) out[b] = sh[0] + fb[0];
}

// ---------------------------------------------------------------------------
extern "C" void kernel_launch(void* const* d_in, const int* in_sizes, int n_in,
                              void* d_out, int out_size, void* d_ws, size_t ws_size,
                              hipStream_t stream) {
  (void)in_sizes; (void)n_in; (void)out_size; (void)ws_size;
  const float* str_fea  = (const float*)d_in[0];
  const int*   comp_fea = (const int*)d_in[1];
  // d_in[2] cell_fea: unused by reference
  const float* atom_table = (const float*)d_in[3];
  const float* comp_W = (const float*)d_in[4];
  const float* comp_b = (const float*)d_in[5];
  const float* pdd_W  = (const float*)d_in[6];
  const float* pdd_b  = (const float*)d_in[7];
  const float* enc_g  = (const float*)d_in[8];
  const float* enc_b  = (const float*)d_in[9];
  const float* sa_qW  = (const float*)d_in[10];
  const float* sa_kW  = (const float*)d_in[11];
  const float* sa_vW  = (const float*)d_in[12];
  const float* sa_vb  = (const float*)d_in[13];
  const float* sa_tW  = (const float*)d_in[14];
  const float* sa_tb  = (const float*)d_in[15];
  const float* sa_bg  = (const float*)d_in[16];
  const float* sa_bb  = (const float*)d_in[17];
  const float* out_W  = (const float*)d_in[18];
  const float* out_b  = (const float*)d_in[19];
  const float* ffn_W  = (const float*)d_in[20];
  const float* ffn_b  = (const float*)d_in[21];
  const float* ln2_g  = (const float*)d_in[22];
  const float* ln2_b  = (const float*)d_in[23];
  const float* fin_W  = (const float*)d_in[24];
  const float* fin_b  = (const float*)d_in[25];

  float* ws = (float*)d_ws;
  const size_t SZ = (size_t)BB * NN * EE;       // 4,194,304 floats
  float* x     = ws;
  float* xinit = ws + 1 * SZ;
  float* h     = ws + 2 * SZ;                   // (B,E,N); reused as xn2 (B,N,E)
  float* x1    = ws + 3 * SZ;
  float* x2    = ws + 4 * SZ;
  float* vbuf  = ws + 5 * SZ;
  float* xrbuf = ws + 6 * SZ;                   // reused as (out1+o2)
  float* o1buf = ws + 7 * SZ;
  float* qbuf  = ws + 8 * SZ;                   // B*N*64
  float* kbuf  = ws + 8 * SZ + SZ / 2;          // B*64*N
  float* csbuf = ws + 9 * SZ;                   // B*N
  float* attn  = ws + 10 * SZ;                  // B*N*N

  embed_kernel<<<BB * NN, EE, 0, stream>>>(str_fea, comp_fea, atom_table,
                                           comp_W, comp_b, pdd_W, pdd_b, x, xinit);

  for (int l = 0; l < 3; ++l) {
    const float* lng = enc_g + l * EE;
    const float* lnb = enc_b + l * EE;
    ln_kernel<true><<<BB * NN, EE, 0, stream>>>(x, lng, lnb, h);

    const float* sain  = h;
    float*       saout = x1;
    for (int hd = 0; hd < 2; ++hd) {
      int wi = l * 2 + hd;
      const float* qW = sa_qW + (size_t)wi * DD2 * EE;
      const float* kW = sa_kW + (size_t)wi * DD2 * EE;
      const float* vW = sa_vW + (size_t)wi * EE * EE;
      const float* vb = sa_vb + (size_t)wi * EE;
      const float* tW = sa_tW + (size_t)wi * EE * EE;
      const float* tb = sa_tb + (size_t)wi * EE;
      const float* bg = sa_bg + (size_t)wi * EE;
      const float* bbv = sa_bb + (size_t)wi * EE;

      gemm_q<<<512, 128, 0, stream>>>(sain, qW, qbuf);
      gemm_proj<64><<<512, 128, 0, stream>>>(sain, kW, nullptr, kbuf);
      gemm_proj<128><<<1024, 128, 0, stream>>>(sain, vW, vb, vbuf);
      gemm_energy<<<4096, 128, 0, stream>>>(qbuf, kbuf, attn);
      softmax_rows<<<BB * NN, 256, 0, stream>>>(attn);
      colsum_kernel<<<BB * 2, 256, 0, stream>>>(attn, csbuf);
      gemm_xr<<<1024, 128, 0, stream>>>(vbuf, attn, csbuf, xrbuf);
      gemm_t<<<1024, 128, 0, stream>>>(sain, xrbuf, tW, tb, bg, bbv, saout);
      sain = x1; saout = x2;
    }

    gemm_out1<<<1024, 128, 0, stream>>>(x1, x2, x, out_W + (size_t)l * 2 * EE * EE,
                                        out_b + l * EE, o1buf);
    ln_kernel<false><<<BB * NN, EE, 0, stream>>>(o1buf, lng, lnb, h);   // xn2 in h
    gemm_ffn<<<1024, 128, 0, stream>>>(h, o1buf, ffn_W + (size_t)l * EE * EE,
                                       ffn_b + l * EE, xrbuf);          // s in xrbuf
    ln_kernel<false><<<BB * NN, EE, 0, stream>>>(xrbuf, lng, lnb, x);
  }

  pool_kernel<<<BB, EE, 0, stream>>>(x, xinit, str_fea, ln2_g, ln2_b,
                                     fin_W, fin_b, (float*)d_out);
}